// InfiniAttention_77146202570759
// MI455X (gfx1250) — compile-verified
//
#include <hip/hip_runtime.h>
#include <hip/hip_bf16.h>

// ---------------------------------------------------------------------------
// InfiniAttention forward for MI455X (gfx1250), fp32 end-to-end.
// All dense math uses V_WMMA_F32_16X16X4_F32 (exact fp32 matrix pipe).
// QKV projection stages its shared A-tile through LDS with the gfx1250
// async global->LDS path (double buffered, s_wait_asynccnt) when available.
// Causal linear attention uses a chunked scan (chunk=16) with the running
// state S[256 x e-block] held in LDS; P/intra/inter/state-update are WMMA.
// ---------------------------------------------------------------------------

typedef __attribute__((ext_vector_type(2))) float v2f;
typedef __attribute__((ext_vector_type(8))) float v8f;
typedef int vi4 __attribute__((vector_size(16)));

#define HID   1024
#define DM    256
#define NB    4
#define SEQL  1024
#define MROWS (NB * SEQL)
#define EPSV  1e-6f

#if __has_builtin(__builtin_amdgcn_global_load_async_to_lds_b128)
#define HAVE_ASYNC_LDS 1
#else
#define HAVE_ASYNC_LDS 0
#endif

#if HAVE_ASYNC_LDS
#if __has_builtin(__builtin_amdgcn_s_wait_asynccnt)
#define WAIT_ASYNC(N) __builtin_amdgcn_s_wait_asynccnt(N)
#else
#define WAIT_ASYNC(N) asm volatile("s_wait_asynccnt " #N ::: "memory")
#endif
#else
#define WAIT_ASYNC(N) ((void)0)
#endif

static __device__ __forceinline__ v8f wmma4(v2f a, v2f b, v8f c) {
  // D = A(16x4 f32) * B(4x16 f32) + C(16x16 f32)
  return __builtin_amdgcn_wmma_f32_16x16x4_f32(
      /*neg_a=*/false, a, /*neg_b=*/false, b,
      /*c_mod=*/(short)0, c, /*reuse_a=*/false, /*reuse_b=*/false);
}

static __device__ __forceinline__ float elu1(float x) {
  return x > 0.f ? x + 1.f : __expf(x);
}

// Copy one 16-byte segment global->LDS; async (ASYNCcnt) when supported.
static __device__ __forceinline__ void copy16(const float* g, float* l) {
#if HAVE_ASYNC_LDS
  __builtin_amdgcn_global_load_async_to_lds_b128(
      (__attribute__((address_space(1))) vi4*)g,
      (__attribute__((address_space(3))) vi4*)l, 0, 0);
#else
  *(float4*)l = *(const float4*)g;
#endif
}

// ---------------------------------------------------------------------------
// Kernel 1: fused QKV projection + elu+1 feature map.
// grid (MROWS/16, 3), block 128 (4 waves). Wave w computes a 16x64 tile.
// The 16x1024 A tile (shared by all 4 waves) is streamed through LDS in
// 16x64 double-buffered chunks (async load-to-LDS); the chunk for step kc+1
// is in flight while WMMA consumes chunk kc. Row stride padded to 68 floats
// so the 16 lanes of an A-fragment read hit 16 distinct LDS banks.
// B[k][n] = w[n*H+k] stays in global (NT gemm -> contiguous 8B k-pairs).
// ---------------------------------------------------------------------------
#define APAD 68
#define KCH  64

__global__ __launch_bounds__(128)
void proj_kernel(const float* __restrict__ hs,
                 const float* __restrict__ wq,
                 const float* __restrict__ wk,
                 const float* __restrict__ wv,
                 float* __restrict__ sq, float* __restrict__ sk,
                 float* __restrict__ vvo) {
  __shared__ float Abuf[2][16 * APAD];

  const int tid  = threadIdx.x;
  const int lane = tid & 31;
  const int wid  = tid >> 5;
  const int hl   = lane >> 4;
  const int l15  = lane & 15;
  const int m0   = blockIdx.x * 16;
  const int mat  = blockIdx.y;                 // 0=q 1=k 2=v
  const float* w = (mat == 0) ? wq : (mat == 1) ? wk : wv;
  float* out     = (mat == 0) ? sq : (mat == 1) ? sk : vvo;
  const int n0   = wid * 64;

  // Stage a 16 x 64 float chunk of A: 256 16B segments, 2 per thread.
  auto stage = [&](int buf, int k0) {
#pragma unroll
    for (int r = 0; r < 2; ++r) {
      const int idx = r * 128 + tid;           // 0..255
      const int row = idx >> 4;                // 16 segments per row
      const int seg = idx & 15;
      copy16(hs + (size_t)(m0 + row) * HID + k0 + seg * 4,
             &Abuf[buf][row * APAD + seg * 4]);
    }
  };

  v8f a0 = {}, a1 = {}, a2 = {}, a3 = {};
  const float* b0 = w + (size_t)(n0 +  0 + l15) * HID + 2 * hl;
  const float* b1 = w + (size_t)(n0 + 16 + l15) * HID + 2 * hl;
  const float* b2 = w + (size_t)(n0 + 32 + l15) * HID + 2 * hl;
  const float* b3 = w + (size_t)(n0 + 48 + l15) * HID + 2 * hl;

  stage(0, 0);
  for (int kc = 0; kc < HID / KCH; ++kc) {
    const int k0 = kc * KCH;
    if (kc + 1 < HID / KCH) {
      stage((kc + 1) & 1, k0 + KCH);
      WAIT_ASYNC(2);                 // chunk kc complete, kc+1 still in flight
    } else {
      WAIT_ASYNC(0);
    }
    __syncthreads();
    const float* al = &Abuf[kc & 1][l15 * APAD + 2 * hl];
#pragma unroll
    for (int kk = 0; kk < KCH; kk += 4) {
      v2f a = *(const v2f*)(al + kk);
      a0 = wmma4(a, *(const v2f*)(b0 + k0 + kk), a0);
      a1 = wmma4(a, *(const v2f*)(b1 + k0 + kk), a1);
      a2 = wmma4(a, *(const v2f*)(b2 + k0 + kk), a2);
      a3 = wmma4(a, *(const v2f*)(b3 + k0 + kk), a3);
    }
    __syncthreads();                 // buffer may be overwritten next round
  }

  const bool act = (mat < 2);
#pragma unroll
  for (int i = 0; i < 8; ++i) {
    const int r = m0 + i + 8 * hl;
    float x0 = a0[i], x1 = a1[i], x2 = a2[i], x3 = a3[i];
    if (act) { x0 = elu1(x0); x1 = elu1(x1); x2 = elu1(x2); x3 = elu1(x3); }
    out[(size_t)r * DM + n0 +  0 + l15] = x0;
    out[(size_t)r * DM + n0 + 16 + l15] = x1;
    out[(size_t)r * DM + n0 + 32 + l15] = x2;
    out[(size_t)r * DM + n0 + 48 + l15] = x3;
  }
}

// ---------------------------------------------------------------------------
// Kernel 2: row norms  normq[m]=sigma_q[m]·mn, normk[m]=sigma_k[m]·mn,
// plus nsum = sum(mn) for the retrieval zero-guard.
// ---------------------------------------------------------------------------
__global__ void rownorm_kernel(const float* __restrict__ sq,
                               const float* __restrict__ sk,
                               const float* __restrict__ mn,
                               float* __restrict__ normq,
                               float* __restrict__ normk,
                               float* __restrict__ nsum) {
  const int m = blockIdx.x * blockDim.x + threadIdx.x;
  if (m < MROWS) {
    float aq = 0.f, ak = 0.f;
    for (int d = 0; d < DM; ++d) {
      const float wv = mn[d];
      aq += sq[(size_t)m * DM + d] * wv;
      ak += sk[(size_t)m * DM + d] * wv;
    }
    normq[m] = aq;
    normk[m] = ak;
  }
  if (blockIdx.x == 0 && threadIdx.x == 0) {
    float s = 0.f;
    for (int d = 0; d < DM; ++d) s += mn[d];
    *nsum = s;
  }
}

// ---------------------------------------------------------------------------
// Kernel 3/5: X @ memory with epilogue.
//  mode 0 (retrieval): out = (nsum<eps ? 0 : acc / max(normX,eps))
//  mode 1 (delta-v)  : out = vbase - acc / max(normX,eps)
// grid (MROWS/16), block 128; wave w -> 16x64 tile, K=256.
// ---------------------------------------------------------------------------
__global__ __launch_bounds__(128)
void xmem_kernel(const float* __restrict__ x, const float* __restrict__ mem,
                 const float* __restrict__ rnorm, const float* __restrict__ nsum,
                 const float* __restrict__ vbase, float* __restrict__ out,
                 int mode) {
  const int lane = threadIdx.x & 31;
  const int wid  = threadIdx.x >> 5;
  const int hl   = lane >> 4;
  const int l15  = lane & 15;
  const int m0   = blockIdx.x * 16;
  const int n0   = wid * 64;

  v8f acc[4] = {{}, {}, {}, {}};
  const float* ar = x + (size_t)(m0 + l15) * DM + 2 * hl;
  for (int k = 0; k < DM; k += 4) {
    v2f a = *(const v2f*)(ar + k);
    const float* r0 = mem + (size_t)(k + 2 * hl) * DM;
    const float* r1 = r0 + DM;
#pragma unroll
    for (int t = 0; t < 4; ++t) {
      const int c = n0 + t * 16 + l15;
      v2f b; b.x = r0[c]; b.y = r1[c];
      acc[t] = wmma4(a, b, acc[t]);
    }
  }
  const float zflag = (*nsum < EPSV) ? 0.f : 1.f;
#pragma unroll
  for (int i = 0; i < 8; ++i) {
    const int r = m0 + i + 8 * hl;
    const float nrm = fmaxf(rnorm[r], EPSV);
#pragma unroll
    for (int t = 0; t < 4; ++t) {
      const int c = n0 + t * 16 + l15;
      const float val = acc[t][i] / nrm;
      if (mode == 0) out[(size_t)r * DM + c] = zflag * val;
      else           out[(size_t)r * DM + c] = vbase[(size_t)r * DM + c] - val;
    }
  }
}

// ---------------------------------------------------------------------------
// Kernel 4: chunked causal linear-attention scan + gate combine.
// grid (NB, DM/32), block 64 (2 waves); each wave owns 16 state columns.
// Per 16-row chunk: P = sq*sk^T (wave 1, WMMA, masked), den via zc (wave 0),
// then each wave: acc = sq@S (inter) + P_masked@v (intra), epilogue divides by
// den and blends with retrieval already staged in `comb`; finally S += sk^T@v.
// ---------------------------------------------------------------------------
__global__ __launch_bounds__(64)
void scan_kernel(const float* __restrict__ sq, const float* __restrict__ sk,
                 const float* __restrict__ vvi, float* __restrict__ comb,
                 const float* __restrict__ gate) {
  __shared__ float Sst[DM][32];    // running state columns for this e-block
  __shared__ float zc[16][DM];     // per-row k-cumsum within chunk
  __shared__ float zrun[DM];       // running k-cumsum
  __shared__ float Pm[16][16];     // masked P
  __shared__ float den[16];

  const int tid  = threadIdx.x;
  const int lane = tid & 31;
  const int wid  = tid >> 5;
  const int hl   = lane >> 4;
  const int l15  = lane & 15;
  const int b    = blockIdx.x;
  const int e0   = blockIdx.y * 32 + wid * 16;   // global e-column base
  const int el   = wid * 16;                     // local column base in Sst

  const float g  = 1.f / (1.f + __expf(-gate[0]));
  const float og = 1.f - g;

  for (int i = tid; i < DM * 32; i += 64) ((float*)Sst)[i] = 0.f;
  for (int d = tid; d < DM; d += 64) zrun[d] = 0.f;
  __syncthreads();

  const int rb0 = b * SEQL;
  for (int c = 0; c < SEQL / 16; ++c) {
    const int rb = rb0 + c * 16;
    __syncthreads();                         // prev chunk fully done
    // Phase A: zc[u][d] = zrun + cumsum of sigma_k within chunk
    for (int d = tid; d < DM; d += 64) {
      float r = zrun[d];
      for (int u = 0; u < 16; ++u) {
        r += sk[(size_t)(rb + u) * DM + d];
        zc[u][d] = r;
      }
      zrun[d] = r;
    }
    __syncthreads();
    // Phase B: wave 1 -> masked P via WMMA; wave 0 lanes 0..15 -> den
    if (wid == 1) {
      v8f p = {};
      const float* ar = sq + (size_t)(rb + l15) * DM + 2 * hl;
      const float* br = sk + (size_t)(rb + l15) * DM + 2 * hl;
      for (int k = 0; k < DM; k += 4)
        p = wmma4(*(const v2f*)(ar + k), *(const v2f*)(br + k), p);
#pragma unroll
      for (int i = 0; i < 8; ++i) {
        const int t = i + 8 * hl, u = l15;
        Pm[t][u] = (u <= t) ? p[i] : 0.f;    // causal mask (u <= t)
      }
    } else if (tid < 16) {
      const float* qr = sq + (size_t)(rb + tid) * DM;
      float s = 0.f;
      for (int d = 0; d < DM; ++d) s += qr[d] * zc[tid][d];
      den[tid] = fmaxf(s, EPSV);
    }
    __syncthreads();
    // Phase C: numerator = sq@S_prev + P_masked@v ; blend into comb
    {
      v8f acc = {};
      const float* ar = sq + (size_t)(rb + l15) * DM + 2 * hl;
      for (int k = 0; k < DM; k += 4) {
        v2f a = *(const v2f*)(ar + k);
        v2f bv; bv.x = Sst[k + 2 * hl][el + l15];
        bv.y = Sst[k + 2 * hl + 1][el + l15];
        acc = wmma4(a, bv, acc);
      }
#pragma unroll
      for (int k = 0; k < 16; k += 4) {
        v2f a; a.x = Pm[l15][k + 2 * hl]; a.y = Pm[l15][k + 2 * hl + 1];
        v2f bv;
        bv.x = vvi[(size_t)(rb + k + 2 * hl) * DM + e0 + l15];
        bv.y = vvi[(size_t)(rb + k + 2 * hl + 1) * DM + e0 + l15];
        acc = wmma4(a, bv, acc);
      }
#pragma unroll
      for (int i = 0; i < 8; ++i) {
        const int t = i + 8 * hl;
        const size_t idx = (size_t)(rb + t) * DM + e0 + l15;
        const float local = acc[i] / den[t];
        comb[idx] = g * comb[idx] + og * local;   // comb held retrieval output
      }
    }
    // Phase D: state update S += sk^T @ v (wave-private columns, no race)
    for (int mt = 0; mt < 16; ++mt) {
      const int d0 = mt * 16;
      v8f cf;
#pragma unroll
      for (int i = 0; i < 8; ++i) cf[i] = Sst[d0 + i + 8 * hl][el + l15];
#pragma unroll
      for (int k = 0; k < 16; k += 4) {
        v2f a;
        a.x = sk[(size_t)(rb + k + 2 * hl) * DM + d0 + l15];
        a.y = sk[(size_t)(rb + k + 2 * hl + 1) * DM + d0 + l15];
        v2f bv;
        bv.x = vvi[(size_t)(rb + k + 2 * hl) * DM + e0 + l15];
        bv.y = vvi[(size_t)(rb + k + 2 * hl + 1) * DM + e0 + l15];
        cf = wmma4(a, bv, cf);
      }
#pragma unroll
      for (int i = 0; i < 8; ++i) Sst[d0 + i + 8 * hl][el + l15] = cf[i];
    }
  }
}

// ---------------------------------------------------------------------------
// Kernel 6: output projection  out[m][h] = sum_d comb[m][d] * w_o[h][d].
// grid (MROWS/16, HID/256), block 128; wave -> 16x64 tile, K=256.
// ---------------------------------------------------------------------------
__global__ __launch_bounds__(128)
void outproj_kernel(const float* __restrict__ comb, const float* __restrict__ wo,
                    float* __restrict__ out) {
  const int lane = threadIdx.x & 31;
  const int wid  = threadIdx.x >> 5;
  const int hl   = lane >> 4;
  const int l15  = lane & 15;
  const int m0   = blockIdx.x * 16;
  const int n0   = blockIdx.y * 256 + wid * 64;

  v8f acc[4] = {{}, {}, {}, {}};
  const float* ar = comb + (size_t)(m0 + l15) * DM + 2 * hl;
  const float* b0 = wo + (size_t)(n0 +  0 + l15) * DM + 2 * hl;
  const float* b1 = wo + (size_t)(n0 + 16 + l15) * DM + 2 * hl;
  const float* b2 = wo + (size_t)(n0 + 32 + l15) * DM + 2 * hl;
  const float* b3 = wo + (size_t)(n0 + 48 + l15) * DM + 2 * hl;
  for (int k = 0; k < DM; k += 4) {
    v2f a = *(const v2f*)(ar + k);
    acc[0] = wmma4(a, *(const v2f*)(b0 + k), acc[0]);
    acc[1] = wmma4(a, *(const v2f*)(b1 + k), acc[1]);
    acc[2] = wmma4(a, *(const v2f*)(b2 + k), acc[2]);
    acc[3] = wmma4(a, *(const v2f*)(b3 + k), acc[3]);
  }
#pragma unroll
  for (int i = 0; i < 8; ++i) {
    const int r = m0 + i + 8 * hl;
#pragma unroll
    for (int t = 0; t < 4; ++t)
      out[(size_t)r * HID + n0 + t * 16 + l15] = acc[t][i];
  }
}

// ---------------------------------------------------------------------------
// Kernel 7: new_memory = memory + (sk^T @ delta_v) / (B*S).
// grid (DM/16), block 128; wave -> 16x64 tile, K = 4096.
// ---------------------------------------------------------------------------
__global__ __launch_bounds__(128)
void memupd_kernel(const float* __restrict__ sk, const float* __restrict__ dv,
                   const float* __restrict__ mem, float* __restrict__ nmem) {
  const int lane = threadIdx.x & 31;
  const int wid  = threadIdx.x >> 5;
  const int hl   = lane >> 4;
  const int l15  = lane & 15;
  const int d0   = blockIdx.x * 16;
  const int n0   = wid * 64;

  v8f acc[4] = {{}, {}, {}, {}};
  for (int k = 0; k < MROWS; k += 4) {
    const int kr = k + 2 * hl;
    v2f a;
    a.x = sk[(size_t)kr * DM + d0 + l15];
    a.y = sk[(size_t)(kr + 1) * DM + d0 + l15];
#pragma unroll
    for (int t = 0; t < 4; ++t) {
      const int c = n0 + t * 16 + l15;
      v2f b;
      b.x = dv[(size_t)kr * DM + c];
      b.y = dv[(size_t)(kr + 1) * DM + c];
      acc[t] = wmma4(a, b, acc[t]);
    }
  }
  const float sc = 1.f / (float)(NB * SEQL);
#pragma unroll
  for (int i = 0; i < 8; ++i) {
    const int r = d0 + i + 8 * hl;
#pragma unroll
    for (int t = 0; t < 4; ++t) {
      const int c = n0 + t * 16 + l15;
      nmem[(size_t)r * DM + c] = mem[(size_t)r * DM + c] + acc[t][i] * sc;
    }
  }
}

// ---------------------------------------------------------------------------
// Kernel 8: new_memory_norm[d] = mn[d] + sum_{b,s} sk / B.
// ---------------------------------------------------------------------------
__global__ void znorm_kernel(const float* __restrict__ sk,
                             const float* __restrict__ mn,
                             float* __restrict__ nnorm) {
  const int d = threadIdx.x;
  float s = 0.f;
  for (int m = 0; m < MROWS; ++m) s += sk[(size_t)m * DM + d];
  nnorm[d] = mn[d] + s / (float)NB;
}

// ---------------------------------------------------------------------------
extern "C" void kernel_launch(void* const* d_in, const int* in_sizes, int n_in,
                              void* d_out, int out_size, void* d_ws,
                              size_t ws_size, hipStream_t stream) {
  (void)in_sizes; (void)n_in; (void)out_size; (void)ws_size;
  const float* hs   = (const float*)d_in[0];
  const float* wq   = (const float*)d_in[1];
  const float* wk   = (const float*)d_in[2];
  const float* wvp  = (const float*)d_in[3];
  const float* wo   = (const float*)d_in[4];
  const float* gate = (const float*)d_in[5];
  const float* mem  = (const float*)d_in[6];
  const float* mn   = (const float*)d_in[7];
  float* out = (float*)d_out;

  float* ws = (float*)d_ws;
  const size_t MD = (size_t)MROWS * DM;      // 1,048,576 floats
  float* sq    = ws;
  float* sk    = ws + MD;
  float* vv    = ws + 2 * MD;
  float* comb  = ws + 3 * MD;                // retrieval, then blended output
  float* dv    = ws + 4 * MD;                // delta_v
  float* normq = ws + 5 * MD;
  float* normk = normq + MROWS;
  float* nsum  = normk + MROWS;              // total ~21 MiB

  proj_kernel<<<dim3(MROWS / 16, 3), 128, 0, stream>>>(hs, wq, wk, wvp,
                                                       sq, sk, vv);
  rownorm_kernel<<<MROWS / 256, 256, 0, stream>>>(sq, sk, mn,
                                                  normq, normk, nsum);
  xmem_kernel<<<MROWS / 16, 128, 0, stream>>>(sq, mem, normq, nsum,
                                              nullptr, comb, 0);
  scan_kernel<<<dim3(NB, DM / 32), 64, 0, stream>>>(sq, sk, vv, comb, gate);
  xmem_kernel<<<MROWS / 16, 128, 0, stream>>>(sk, mem, normk, nsum,
                                              vv, dv, 1);
  outproj_kernel<<<dim3(MROWS / 16, HID / 256), 128, 0, stream>>>(comb, wo,
                                                                  out);
  float* nmem = out + (size_t)MROWS * HID;
  memupd_kernel<<<DM / 16, 128, 0, stream>>>(sk, dv, mem, nmem);
  znorm_kernel<<<1, DM, 0, stream>>>(sk, mn, nmem + DM * DM);
}